// WSDDN_21234318311646
// MI455X (gfx1250) — compile-verified
//
#include <hip/hip_runtime.h>
#include <float.h>
#include <stdint.h>

typedef __attribute__((ext_vector_type(2))) float v2f;
typedef __attribute__((ext_vector_type(8))) float v8f;
typedef __attribute__((ext_vector_type(4))) int   v4i;

#define R_ROIS 500
#define MPAD   512
#define C_CH   512
#define H_F    38
#define W_F    38
#define KCLS   20
#define HID    4096
#define FEAT   4608   // 512*9

// CDNA5 async global->LDS path (tracked on ASYNCcnt), guarded so the file
// compiles on toolchains without the builtin.
#if defined(__HIP_DEVICE_COMPILE__) && __has_builtin(__builtin_amdgcn_global_load_async_to_lds_b128)
#define HAVE_ASYNC_LDS 1
#else
#define HAVE_ASYNC_LDS 0
#endif

#if HAVE_ASYNC_LDS
typedef __attribute__((address_space(1))) v4i* gv4i_ptr;   // printed as "int __vector(4) __device__ *"
typedef __attribute__((address_space(3))) v4i* lv4i_ptr;   // LDS

__device__ inline void wait_async0() {
#if __has_builtin(__builtin_amdgcn_s_wait_asynccnt)
    __builtin_amdgcn_s_wait_asynccnt(0);
#else
    asm volatile("s_wait_asynccnt 0x0" ::: "memory");
#endif
}
__device__ inline void async_load_b128(const float* g, float* lds) {
    __builtin_amdgcn_global_load_async_to_lds_b128(
        (gv4i_ptr)(uintptr_t)g,      // AS(1): global addresses == generic values
        (lv4i_ptr)lds,               // addrspacecast generic -> LDS
        0, 0);
}
#endif

// ---------------- ROI max pool ----------------
__global__ void roi_pool_kernel(const float* __restrict__ fmap,
                                const float* __restrict__ rois,
                                float* __restrict__ feat,
                                float* __restrict__ rois_sc) {
    int r = blockIdx.x;
    int t = threadIdx.x;
    if (r >= R_ROIS) {             // zero M-padding rows for the GEMM
        for (int i = t; i < FEAT; i += 256) feat[(size_t)r * FEAT + i] = 0.f;
        return;
    }
    float fx1 = floorf(rois[r*4+0] * (1.f/16.f));
    float fy1 = floorf(rois[r*4+1] * (1.f/16.f));
    float fx2 = floorf(rois[r*4+2] * (1.f/16.f));
    float fy2 = floorf(rois[r*4+3] * (1.f/16.f));
    if (t == 0) { rois_sc[r*4+0]=fx1; rois_sc[r*4+1]=fy1; rois_sc[r*4+2]=fx2; rois_sc[r*4+3]=fy2; }
    int x1 = (int)fx1, y1 = (int)fy1, x2 = (int)fx2, y2 = (int)fy2;
    int Lx = x2 - x1, Ly = y2 - y1;   // both >= 3 given input ranges
    for (int idx = t; idx < FEAT; idx += 256) {
        int c  = idx / 9;
        int b  = idx % 9;
        int ry = b / 3, rx = b % 3;
        int sx = x1 + (rx * Lx) / 3;
        int ex = x1 + ((rx + 1) * Lx + 2) / 3;   // ceil
        int sy = y1 + (ry * Ly) / 3;
        int ey = y1 + ((ry + 1) * Ly + 2) / 3;
        const float* fc = fmap + (size_t)c * (H_F * W_F);
        float m = -FLT_MAX;
        for (int y = sy; y < ey; ++y)
            for (int x = sx; x < ex; ++x)
                m = fmaxf(m, fc[y * W_F + x]);
        feat[(size_t)r * FEAT + idx] = m;
    }
}

// ---------------- fp32 WMMA GEMM: C[M,N] = A[M,K] @ B[K,N] + bias ----------------
// Double-buffered LDS, software-pipelined staging; B tile staged with
// GLOBAL_LOAD_ASYNC_TO_LDS_B128 when available (ASYNCcnt), A tile through
// registers (needs k-major transpose, no fp32 TR load exists).
#define BM 64
#define BN 128
#define KC 16

__global__ __launch_bounds__(256) void gemm_wmma_f32(
        const float* __restrict__ A, const float* __restrict__ B,
        const float* __restrict__ bias, float* __restrict__ C,
        int N, int K) {
    __shared__ float As[2][KC][BM + 1];   // k-major A tile
    __shared__ float Bs[2][KC][BN + 4];

    const int tid  = threadIdx.x;
    const int row0 = blockIdx.y * BM;
    const int col0 = blockIdx.x * BN;

    const int wave   = tid >> 5;
    const int lane   = tid & 31;
    const int half   = lane >> 4;     // K half select per ISA A/B layout
    const int lm     = lane & 15;
    const int m_tile = wave & 3;      // 4 M-tiles of 16
    const int n_grp  = wave >> 2;     // 2 N-groups of 64

    v8f c0 = {}, c1 = {}, c2 = {}, c3 = {};

    const int am  = tid & 63, ak4 = tid >> 6;   // A loader: 64 rows x 4 k4-chunks
    const int bn4 = tid & 31, bk  = tid >> 5;   // B loader: 32 col4-chunks x 8 rows (x2)

    const int nchunks = K / KC;

    // ---- prologue: stage chunk 0
    float4 av = *(const float4*)(A + (size_t)(row0 + am) * K + ak4 * 4);
#if HAVE_ASYNC_LDS
    async_load_b128(B + (size_t)(bk    ) * N + col0 + bn4 * 4, &Bs[0][bk    ][bn4 * 4]);
    async_load_b128(B + (size_t)(bk + 8) * N + col0 + bn4 * 4, &Bs[0][bk + 8][bn4 * 4]);
#else
    float4 bv0 = *(const float4*)(B + (size_t)(bk    ) * N + col0 + bn4 * 4);
    float4 bv1 = *(const float4*)(B + (size_t)(bk + 8) * N + col0 + bn4 * 4);
#endif

    for (int i = 0; i < nchunks; ++i) {
        const int cur = i & 1;
        const int nxt = cur ^ 1;

#if HAVE_ASYNC_LDS
        wait_async0();                       // this wave's B tile landed in LDS
#endif
        As[cur][ak4*4+0][am] = av.x; As[cur][ak4*4+1][am] = av.y;
        As[cur][ak4*4+2][am] = av.z; As[cur][ak4*4+3][am] = av.w;
#if !HAVE_ASYNC_LDS
        *(float4*)&Bs[cur][bk    ][bn4 * 4] = bv0;
        *(float4*)&Bs[cur][bk + 8][bn4 * 4] = bv1;
#endif
        __syncthreads();                     // (compiler waits dscnt before barrier)

        if (i + 1 < nchunks) {               // prefetch next chunk, overlapped with WMMAs
            const int k0n = (i + 1) * KC;
            av = *(const float4*)(A + (size_t)(row0 + am) * K + k0n + ak4 * 4);
#if HAVE_ASYNC_LDS
            async_load_b128(B + (size_t)(k0n + bk    ) * N + col0 + bn4 * 4, &Bs[nxt][bk    ][bn4 * 4]);
            async_load_b128(B + (size_t)(k0n + bk + 8) * N + col0 + bn4 * 4, &Bs[nxt][bk + 8][bn4 * 4]);
#else
            bv0 = *(const float4*)(B + (size_t)(k0n + bk    ) * N + col0 + bn4 * 4);
            bv1 = *(const float4*)(B + (size_t)(k0n + bk + 8) * N + col0 + bn4 * 4);
#endif
        }

        #pragma unroll
        for (int kk = 0; kk < KC; kk += 4) {
            v2f a;
            a.x = As[cur][kk + 2*half    ][m_tile * 16 + lm];
            a.y = As[cur][kk + 2*half + 1][m_tile * 16 + lm];
            v2f b0, b1, b2, b3;
            const int nb = n_grp * 64 + lm;
            b0.x = Bs[cur][kk + 2*half][nb +  0]; b0.y = Bs[cur][kk + 2*half + 1][nb +  0];
            b1.x = Bs[cur][kk + 2*half][nb + 16]; b1.y = Bs[cur][kk + 2*half + 1][nb + 16];
            b2.x = Bs[cur][kk + 2*half][nb + 32]; b2.y = Bs[cur][kk + 2*half + 1][nb + 32];
            b3.x = Bs[cur][kk + 2*half][nb + 48]; b3.y = Bs[cur][kk + 2*half + 1][nb + 48];
            c0 = __builtin_amdgcn_wmma_f32_16x16x4_f32(false, a, false, b0, (short)0, c0, false, false);
            c1 = __builtin_amdgcn_wmma_f32_16x16x4_f32(false, a, false, b1, (short)0, c1, false, false);
            c2 = __builtin_amdgcn_wmma_f32_16x16x4_f32(false, a, false, b2, (short)0, c2, false, false);
            c3 = __builtin_amdgcn_wmma_f32_16x16x4_f32(false, a, false, b3, (short)0, c3, false, false);
        }
        // no trailing barrier needed: buffer 'nxt' writes above happen after this
        // iteration's barrier, and all reads of 'nxt' (previous iteration) were
        // dscnt-consumed before their owners' WMMAs preceding that barrier.
    }

    // Epilogue: C/D layout -> lanes 0-15: M=r, lanes 16-31: M=8+r; N = lm
    const int ocol = col0 + n_grp * 64 + lm;
    #pragma unroll
    for (int r = 0; r < 8; ++r) {
        int orow = row0 + m_tile * 16 + half * 8 + r;
        float* cp = C + (size_t)orow * N;
        cp[ocol +  0] = c0[r] + bias[ocol +  0];
        cp[ocol + 16] = c1[r] + bias[ocol + 16];
        cp[ocol + 32] = c2[r] + bias[ocol + 32];
        cp[ocol + 48] = c3[r] + bias[ocol + 48];
    }
}

// ---------------- small heads: logits[r,k] = fc7[r,:] @ W[:,k] + b[k] ----------------
__device__ inline float wave_sum(float v) {
    #pragma unroll
    for (int off = 16; off > 0; off >>= 1) v += __shfl_down(v, off, 32);
    return v;
}

__global__ void head_kernel(const float* __restrict__ fc7, const float* __restrict__ W,
                            const float* __restrict__ b, float* __restrict__ out) {
    int r = blockIdx.x, t = threadIdx.x;
    float acc[KCLS];
    #pragma unroll
    for (int k = 0; k < KCLS; ++k) acc[k] = 0.f;
    const float* row = fc7 + (size_t)r * HID;
    for (int j = t; j < HID; j += 256) {
        float v = row[j];
        const float* w = W + (size_t)j * KCLS;
        #pragma unroll
        for (int k = 0; k < KCLS; ++k) acc[k] += v * w[k];
    }
    __shared__ float red[8][KCLS];
    int lane = t & 31, wv = t >> 5;
    #pragma unroll
    for (int k = 0; k < KCLS; ++k) {
        float s = wave_sum(acc[k]);
        if (lane == 0) red[wv][k] = s;
    }
    __syncthreads();
    if (t < KCLS) {
        float s = 0.f;
        for (int i = 0; i < 8; ++i) s += red[i][t];
        out[r * KCLS + t] = s + b[t];
    }
}

// ---------------- row softmax (pc) ----------------
__global__ void pc_kernel(const float* __restrict__ lc, float* __restrict__ pc) {
    int r = blockIdx.x, t = threadIdx.x;
    float v = (t < KCLS) ? lc[r * KCLS + t] : -FLT_MAX;
    float m = v;
    #pragma unroll
    for (int off = 16; off > 0; off >>= 1) m = fmaxf(m, __shfl_down(m, off, 32));
    m = __shfl(m, 0, 32);
    float e = (t < KCLS) ? expf(v - m) : 0.f;
    float s = wave_sum(e);
    s = __shfl(s, 0, 32);
    if (t < KCLS) pc[r * KCLS + t] = e / s;
}

// ---------------- column softmax (pd), scores, per-class sums ----------------
__global__ void pd_scores_kernel(const float* __restrict__ ld, const float* __restrict__ pc,
                                 float* __restrict__ scores, float* __restrict__ outsum) {
    int k = blockIdx.x, t = threadIdx.x;
    __shared__ float red[256];
    float m = -FLT_MAX;
    for (int r = t; r < R_ROIS; r += 256) m = fmaxf(m, ld[r * KCLS + k]);
    red[t] = m; __syncthreads();
    for (int s = 128; s > 0; s >>= 1) { if (t < s) red[t] = fmaxf(red[t], red[t+s]); __syncthreads(); }
    m = red[0]; __syncthreads();
    float sum = 0.f;
    for (int r = t; r < R_ROIS; r += 256) sum += expf(ld[r * KCLS + k] - m);
    red[t] = sum; __syncthreads();
    for (int s = 128; s > 0; s >>= 1) { if (t < s) red[t] += red[t+s]; __syncthreads(); }
    float S = red[0]; __syncthreads();
    float osum = 0.f;
    for (int r = t; r < R_ROIS; r += 256) {
        float sc = pc[r * KCLS + k] * expf(ld[r * KCLS + k] - m) / S;
        scores[r * KCLS + k] = sc;
        osum += sc;
    }
    red[t] = osum; __syncthreads();
    for (int s = 128; s > 0; s >>= 1) { if (t < s) red[t] += red[t+s]; __syncthreads(); }
    if (t == 0) outsum[k] = red[0];
}

// ---------------- BCE ----------------
__global__ void bce_kernel(const float* __restrict__ outsum, const float* __restrict__ label,
                           float* __restrict__ bce) {
    int t = threadIdx.x;
    float v = 0.f;
    if (t < KCLS) {
        float o = outsum[t], l = label[t];
        v = l * logf(o) + (1.f - l) * logf(1.f - o);
    }
    v = wave_sum(v);
    if (t == 0) bce[0] = -v / (float)KCLS;
}

// ---------------- spatial regularizer (one block per class) ----------------
__global__ void reg_kernel(const float* __restrict__ scores, const float* __restrict__ fc7,
                           const float* __restrict__ rois_sc, const float* __restrict__ label,
                           float* __restrict__ regk) {
    int k = blockIdx.x, t = threadIdx.x;
    __shared__ float red[256];
    __shared__ int   redi[256];
    __shared__ int   h_s;
    __shared__ float hh_s;

    // stable argmin over scores[:,k]  (== argsort(...)[0])
    float mv = FLT_MAX; int mi = R_ROIS;
    for (int r = t; r < R_ROIS; r += 256) {
        float s = scores[r * KCLS + k];
        if (s < mv) { mv = s; mi = r; }
    }
    red[t] = mv; redi[t] = mi; __syncthreads();
    for (int s = 128; s > 0; s >>= 1) {
        if (t < s) {
            if (red[t+s] < red[t] || (red[t+s] == red[t] && redi[t+s] < redi[t])) {
                red[t] = red[t+s]; redi[t] = redi[t+s];
            }
        }
        __syncthreads();
    }
    if (t == 0) h_s = redi[0];
    __syncthreads();
    const int h = h_s;
    const float* fh = fc7 + (size_t)h * HID;

    // ||fc7[h]||^2 (contribution of every iou<=0.6 box)
    float p = 0.f;
    for (int j = t; j < HID; j += 256) { float b = fh[j]; p += b * b; }
    red[t] = p; __syncthreads();
    for (int s = 128; s > 0; s >>= 1) { if (t < s) red[t] += red[t+s]; __syncthreads(); }
    if (t == 0) hh_s = red[0];
    __syncthreads();
    const float hh = hh_s;

    const float hx1 = rois_sc[h*4+0], hy1 = rois_sc[h*4+1];
    const float hx2 = rois_sc[h*4+2], hy2 = rois_sc[h*4+3];
    const float b2x2 = hx1 + hx2, b2y2 = hy1 + hy2;
    const float a2 = (hx2 + 1.f) * (hy2 + 1.f);

    float total = 0.f;
    for (int r = 0; r < R_ROIS; ++r) {
        if (r == h) continue;
        float sk = scores[r * KCLS + k];
        float s2 = sk * sk;
        float x1 = rois_sc[r*4+0], y1 = rois_sc[r*4+1];
        float x2 = rois_sc[r*4+2], y2 = rois_sc[r*4+3];
        float b1x2 = x1 + x2, b1y2 = y1 + y2;
        float iw = fmaxf(fminf(b1x2, b2x2) - fmaxf(x1, hx1) + 1.f, 0.f);
        float ih = fmaxf(fminf(b1y2, b2y2) - fmaxf(y1, hy1) + 1.f, 0.f);
        float inter = iw * ih;
        float a1 = (x2 + 1.f) * (y2 + 1.f);
        float iou = inter / (a1 + a2 - inter);
        if (iou > 0.6f) {
            const float* fr = fc7 + (size_t)r * HID;
            float acc = 0.f;
            for (int j = t; j < HID; j += 256) { float d = fr[j] - fh[j]; acc += d * d; }
            total += s2 * acc;
        } else if (t == 0) {
            total += s2 * hh;    // mask==0: diff = -fc7[h]*s
        }
    }
    red[t] = total; __syncthreads();
    for (int s = 128; s > 0; s >>= 1) { if (t < s) red[t] += red[t+s]; __syncthreads(); }
    if (t == 0) regk[k] = label[k] * 0.5f * red[0];
}

// ---------------- finalize ----------------
__global__ void finalize_kernel(const float* __restrict__ regk, const float* __restrict__ bce,
                                float* __restrict__ out) {
    if (threadIdx.x == 0 && blockIdx.x == 0) {
        float reg = 0.f;
        for (int k = 0; k < KCLS; ++k) reg += regk[k];
        reg /= (float)KCLS;
        out[10000] = bce[0] + reg;
        out[10001] = reg;
    }
}

extern "C" void kernel_launch(void* const* d_in, const int* in_sizes, int n_in,
                              void* d_out, int out_size, void* d_ws, size_t ws_size,
                              hipStream_t stream) {
    const float* fmap  = (const float*)d_in[0];
    const float* rois  = (const float*)d_in[1];
    const float* label = (const float*)d_in[2];
    const float* W6    = (const float*)d_in[3];
    const float* b6    = (const float*)d_in[4];
    const float* W7    = (const float*)d_in[5];
    const float* b7    = (const float*)d_in[6];
    const float* W8c   = (const float*)d_in[7];
    const float* b8c   = (const float*)d_in[8];
    const float* W8d   = (const float*)d_in[9];
    const float* b8d   = (const float*)d_in[10];
    float* out = (float*)d_out;
    float* ws  = (float*)d_ws;

    // workspace layout (floats)
    float* feat    = ws;                               // 512*4608
    float* rois_sc = feat    + (size_t)MPAD * FEAT;    // 2048
    float* fc6     = rois_sc + 2048;                   // 512*4096
    float* fc7     = fc6     + (size_t)MPAD * HID;     // 512*4096
    float* lc      = fc7     + (size_t)MPAD * HID;     // 10240
    float* ld      = lc      + 10240;                  // 10240
    float* pc      = ld      + 10240;                  // 10240
    float* outsum  = pc      + 10240;                  // 32
    float* bce     = outsum  + 32;                     // 32
    float* regk    = bce     + 32;                     // 32

    roi_pool_kernel<<<MPAD, 256, 0, stream>>>(fmap, rois, feat, rois_sc);
    gemm_wmma_f32<<<dim3(HID / BN, MPAD / BM), 256, 0, stream>>>(feat, W6, b6, fc6, HID, FEAT);
    gemm_wmma_f32<<<dim3(HID / BN, MPAD / BM), 256, 0, stream>>>(fc6,  W7, b7, fc7, HID, HID);
    head_kernel<<<R_ROIS, 256, 0, stream>>>(fc7, W8c, b8c, lc);
    head_kernel<<<R_ROIS, 256, 0, stream>>>(fc7, W8d, b8d, ld);
    pc_kernel<<<R_ROIS, 32, 0, stream>>>(lc, pc);
    pd_scores_kernel<<<KCLS, 256, 0, stream>>>(ld, pc, out, outsum);
    bce_kernel<<<1, 32, 0, stream>>>(outsum, label, bce);
    reg_kernel<<<KCLS, 256, 0, stream>>>(out, fc7, rois_sc, label, regk);
    finalize_kernel<<<1, 1, 0, stream>>>(regk, bce, out);
}